// SelfAttention_10634339025419
// MI455X (gfx1250) — compile-verified
//
#include <hip/hip_runtime.h>

typedef float v2f __attribute__((ext_vector_type(2)));
typedef float v8f __attribute__((ext_vector_type(8)));

#define NBATCH 4
#define NSEQ   2048
#define DDIM   1024

// ---------------------------------------------------------------------------
// Kernel 1: fused QKV projection.  out[z][row][e] = sum_d X[row][d]*W_z[e][d] + b_z[e]
// grid = (DDIM/64 e-tiles, 8192/64 row-tiles, 3), block = 256 (8 waves)
// Each wave computes a 16x32 piece (two 16x16 WMMA tiles) of the 64x64 WG tile.
// ---------------------------------------------------------------------------
__global__ __launch_bounds__(256)
void qkv_proj_kernel(const float* __restrict__ X,
                     const float* __restrict__ Wq, const float* __restrict__ bq,
                     const float* __restrict__ Wk, const float* __restrict__ bk,
                     const float* __restrict__ Wv, const float* __restrict__ bv,
                     float* __restrict__ out_qkv)
{
    __shared__ float Xs[64][36];   // stride 36 dwords: 16B-aligned float4, conflict-free
    __shared__ float Ws[64][36];

    const int z = blockIdx.z;
    const float* W    = (z == 0) ? Wq : (z == 1) ? Wk : Wv;
    const float* bias = (z == 0) ? bq : (z == 1) ? bk : bv;
    float* out = out_qkv + (size_t)z * ((size_t)NBATCH * NSEQ * DDIM);

    const int ebase = blockIdx.x * 64;
    const int rbase = blockIdx.y * 64;

    const int tid  = threadIdx.x;
    const int lane = tid & 31;
    const int wave = tid >> 5;

    const int wr = (wave & 3) * 16;   // wave's row block inside 64
    const int wc = (wave >> 2) * 32;  // wave's col block inside 64 (2 tiles of 16)

    v8f acc0 = {};
    v8f acc1 = {};

    const int lr = tid >> 3;          // 0..31
    const int lc = (tid & 7) * 4;     // 0..28 step 4

    const int m  = lane & 15;
    const int kb = (lane >> 4) * 2;

    for (int kc = 0; kc < DDIM; kc += 32) {
        *(float4*)&Xs[lr][lc]      = *(const float4*)&X[(size_t)(rbase + lr)      * DDIM + kc + lc];
        *(float4*)&Xs[lr + 32][lc] = *(const float4*)&X[(size_t)(rbase + lr + 32) * DDIM + kc + lc];
        *(float4*)&Ws[lr][lc]      = *(const float4*)&W[(size_t)(ebase + lr)      * DDIM + kc + lc];
        *(float4*)&Ws[lr + 32][lc] = *(const float4*)&W[(size_t)(ebase + lr + 32) * DDIM + kc + lc];
        __syncthreads();

        #pragma unroll
        for (int kk = 0; kk < 32; kk += 4) {
            v2f a  = *(const v2f*)&Xs[wr + m][kk + kb];        // A[m][kb..kb+1]
            v2f b0 = *(const v2f*)&Ws[wc + m][kk + kb];        // B[kb..kb+1][n], n=lane&15
            v2f b1 = *(const v2f*)&Ws[wc + 16 + m][kk + kb];
            acc0 = __builtin_amdgcn_wmma_f32_16x16x4_f32(false, a, false, b0, (short)0, acc0, false, false);
            acc1 = __builtin_amdgcn_wmma_f32_16x16x4_f32(false, a, false, b1, (short)0, acc1, false, false);
        }
        __syncthreads();
    }

    // Epilogue: bias + store.  D layout: vgpr i -> row = i + 8*(lane>>4), col = lane&15.
    const int mo = 8 * (lane >> 4);
    const float bv0 = bias[ebase + wc + m];
    const float bv1 = bias[ebase + wc + 16 + m];
    #pragma unroll
    for (int i = 0; i < 8; ++i) {
        const int row = rbase + wr + i + mo;
        out[(size_t)row * DDIM + ebase + wc + m]      = acc0[i] + bv0;
        out[(size_t)row * DDIM + ebase + wc + 16 + m] = acc1[i] + bv1;
    }
}

// ---------------------------------------------------------------------------
// Kernel 2: causal (tril, NO softmax) attention:  O = tril(Q Kt * scale) V
// grid = (NSEQ/32 q-blocks, NBATCH), block = 256 (8 waves).
// Per k-block of 64:
//   Phase A: S(32x64) = Q Kt, one 16x16 tile per wave, D staged in LDS chunks.
//   Phase B: each wave accumulates O tile 32 q-rows x 128 d-cols (2x8 v8f accs),
//            so each V B-fragment feeds 2 WMMAs and each V element is loaded by
//            exactly one wave (halves L2 traffic vs 16x256 tiling).
// ---------------------------------------------------------------------------
__global__ __launch_bounds__(256)
void causal_attn_kernel(const float* __restrict__ QKV, float* __restrict__ outp)
{
    const float* Q = QKV;
    const float* K = QKV + (size_t)1 * NBATCH * NSEQ * DDIM;
    const float* V = QKV + (size_t)2 * NBATCH * NSEQ * DDIM;

    const int b      = blockIdx.y;
    const int qstart = blockIdx.x * 32;
    const int qend   = qstart + 31;

    const float* Qb = Q + (size_t)b * NSEQ * DDIM;
    const float* Kb = K + (size_t)b * NSEQ * DDIM;
    const float* Vb = V + (size_t)b * NSEQ * DDIM;
    float*       Ob = outp + (size_t)b * NSEQ * DDIM;

    __shared__ float Qs[32][36];
    __shared__ float Ks[64][36];
    __shared__ float Ss[32][68];

    const int tid  = threadIdx.x;
    const int lane = tid & 31;
    const int wave = tid >> 5;

    const int s_qr = (wave & 1) * 16;   // S phase: q row block
    const int s_kc = (wave >> 1) * 16;  // S phase: k col block (4 of them)
    const int o_d  = wave * 128;        // O phase: d range (8 x 128 = 1024)

    const int m  = lane & 15;
    const int kb = (lane >> 4) * 2;
    const int mo = 8 * (lane >> 4);

    const int lr = tid >> 3;        // 0..31
    const int lc = (tid & 7) * 4;   // 0..28

    const float scale = 0.03125f;   // 1/sqrt(1024)

    v8f oacc[2][8];                 // [q-row block 0/16][d tile]
    #pragma unroll
    for (int r = 0; r < 2; ++r)
        #pragma unroll
        for (int t = 0; t < 8; ++t) oacc[r][t] = (v8f){};

    for (int kstart = 0; kstart <= qend; kstart += 64) {
        // ---- Phase A: S = Q * Kt over full D ----
        v8f sacc = {};
        for (int dc = 0; dc < DDIM; dc += 32) {
            *(float4*)&Qs[lr][lc]      = *(const float4*)&Qb[(size_t)(qstart + lr)      * DDIM + dc + lc];
            *(float4*)&Ks[lr][lc]      = *(const float4*)&Kb[(size_t)(kstart + lr)      * DDIM + dc + lc];
            *(float4*)&Ks[lr + 32][lc] = *(const float4*)&Kb[(size_t)(kstart + lr + 32) * DDIM + dc + lc];
            __syncthreads();

            #pragma unroll
            for (int kk = 0; kk < 32; kk += 4) {
                v2f a  = *(const v2f*)&Qs[s_qr + m][kk + kb];   // A[m][d..d+1]
                v2f bb = *(const v2f*)&Ks[s_kc + m][kk + kb];   // B[d][n] = K[n][d]
                sacc = __builtin_amdgcn_wmma_f32_16x16x4_f32(false, a, false, bb, (short)0, sacc, false, false);
            }
            __syncthreads();
        }

        // scale + causal mask in registers, spill S tile to LDS
        #pragma unroll
        for (int i = 0; i < 8; ++i) {
            const int qrow = qstart + s_qr + i + mo;
            const int kcol = kstart + s_kc + m;
            float s = sacc[i] * scale;
            if (kcol > qrow) s = 0.0f;
            Ss[s_qr + i + mo][s_kc + m] = s;
        }
        __syncthreads();

        // ---- Phase B: O += S * V ----
        #pragma unroll
        for (int kk = 0; kk < 64; kk += 4) {
            const v2f a0 = *(const v2f*)&Ss[m][kk + kb];        // q rows 0..15
            const v2f a1 = *(const v2f*)&Ss[16 + m][kk + kb];   // q rows 16..31
            const float* vrow = &Vb[(size_t)(kstart + kk + kb) * DDIM];
            #pragma unroll
            for (int t = 0; t < 8; ++t) {
                const int dcol = o_d + t * 16 + m;              // n = lane&15
                v2f bb;
                bb[0] = vrow[dcol];                             // V[k  ][dcol]
                bb[1] = vrow[DDIM + dcol];                      // V[k+1][dcol]
                oacc[0][t] = __builtin_amdgcn_wmma_f32_16x16x4_f32(false, a0, false, bb, (short)0, oacc[0][t], false, false);
                oacc[1][t] = __builtin_amdgcn_wmma_f32_16x16x4_f32(false, a1, false, bb, (short)0, oacc[1][t], false, false);
            }
        }
        // staging barriers at the top of the next iteration fence Ss reuse
    }

    // ---- store O ----
    #pragma unroll
    for (int r = 0; r < 2; ++r) {
        #pragma unroll
        for (int t = 0; t < 8; ++t) {
            const int dcol = o_d + t * 16 + m;
            #pragma unroll
            for (int i = 0; i < 8; ++i) {
                const int qrow = qstart + r * 16 + i + mo;
                Ob[(size_t)qrow * DDIM + dcol] = oacc[r][t][i];
            }
        }
    }
}

// ---------------------------------------------------------------------------
extern "C" void kernel_launch(void* const* d_in, const int* in_sizes, int n_in,
                              void* d_out, int out_size, void* d_ws, size_t ws_size,
                              hipStream_t stream) {
    const float* X  = (const float*)d_in[0];
    const float* Wq = (const float*)d_in[1];
    const float* bq = (const float*)d_in[2];
    const float* Wk = (const float*)d_in[3];
    const float* bk = (const float*)d_in[4];
    const float* Wv = (const float*)d_in[5];
    const float* bv = (const float*)d_in[6];

    float* qkv = (float*)d_ws;  // [3][4][2048][1024] f32 = 100.7 MB scratch

    dim3 g1(DDIM / 64, (NBATCH * NSEQ) / 64, 3);
    qkv_proj_kernel<<<g1, 256, 0, stream>>>(X, Wq, bq, Wk, bk, Wv, bv, qkv);

    dim3 g2(NSEQ / 32, NBATCH);
    causal_attn_kernel<<<g2, 256, 0, stream>>>(qkv, (float*)d_out);
}